// GCN_86268713107997
// MI455X (gfx1250) — compile-verified
//
#include <hip/hip_runtime.h>
#include <hip/hip_bf16.h>

typedef __attribute__((ext_vector_type(2))) float v2f;
typedef __attribute__((ext_vector_type(8))) float v8f;
typedef __attribute__((ext_vector_type(4))) unsigned int v4u;
typedef __attribute__((ext_vector_type(8))) int v8i;
typedef __attribute__((ext_vector_type(4))) int v4i;

#define DIM 64
#define ROWS_PER_BLOCK 128   // 8 waves x 16 rows

// ---------------------------------------------------------------------------
// Degree / normalization kernels
// ---------------------------------------------------------------------------
__global__ void deg_init_kernel(float* __restrict__ deg, int n) {
    int i = blockIdx.x * blockDim.x + threadIdx.x;
    if (i < n) deg[i] = 1.0f;  // self-loop contributes 1 to every node's degree
}

__global__ void deg_count_kernel(const int* __restrict__ dst, float* __restrict__ deg, int ne) {
    int e = blockIdx.x * blockDim.x + threadIdx.x;
    if (e < ne) atomicAdd(&deg[dst[e]], 1.0f);
}

__global__ void deg_rsqrt_kernel(const float* __restrict__ deg, float* __restrict__ dinv, int n) {
    int i = blockIdx.x * blockDim.x + threadIdx.x;
    if (i < n) dinv[i] = rsqrtf(deg[i]);  // deg >= 1 always (self-loops)
}

// ---------------------------------------------------------------------------
// Tensor Data Mover: 2D tile load Global -> LDS.
// D# per CDNA5 ISA 8.3/8.4:
//  group0: [1:0]=count=1, [63:32]=lds_addr, [120:64]=global_addr, [127:126]=type=2
//  group1: [17:16]=data_size (2 -> 4B), [79:48]=tensor_dim0, [111:80]=tensor_dim1,
//          [127:112]=tile_dim0, [143:128]=tile_dim1, [207:160]=tensor_dim0_stride
//  groups 2/3 (+ext): zero (2-D tensor, no dims 2..4)
// clang-23 toolchain: 6-arg builtin (v4u, v8i, v4i, v4i, v8i, i32 cpol)
// ---------------------------------------------------------------------------
__device__ __forceinline__ void tdm_load_2d(const void* gptr, unsigned lds_off,
                                            unsigned dim0, unsigned dim1,
                                            unsigned tile0, unsigned tile1,
                                            unsigned stride0) {
    unsigned long long ga = (unsigned long long)gptr;
    v4u g0;
    g0.x = 0x1u;                                              // count=1, user-mode
    g0.y = lds_off;                                           // LDS byte address
    g0.z = (unsigned)(ga & 0xffffffffu);                      // global_addr[31:0]
    g0.w = (unsigned)((ga >> 32) & 0x1ffffffu) | (2u << 30);  // addr[56:32] | type=2
    v8i g1;
    g1[0] = (int)(2u << 16);                                  // data_size=2 (4 bytes)
    g1[1] = (int)((dim0 & 0xffffu) << 16);                    // tensor_dim0 lo16
    g1[2] = (int)(((dim0 >> 16) & 0xffffu) | ((dim1 & 0xffffu) << 16));
    g1[3] = (int)(((dim1 >> 16) & 0xffffu) | ((tile0 & 0xffffu) << 16));
    g1[4] = (int)(tile1 & 0xffffu);                           // tile_dim1 (tile_dim2=0)
    g1[5] = (int)stride0;                                     // tensor_dim0_stride lo32
    g1[6] = 0;
    g1[7] = 0;
    v4i g2 = {0, 0, 0, 0};
    v4i g3 = {0, 0, 0, 0};
    v8i g4 = {0, 0, 0, 0, 0, 0, 0, 0};
    __builtin_amdgcn_tensor_load_to_lds(g0, g1, g2, g3, g4, 0);
}

__device__ __forceinline__ unsigned lds_offset(const void* p) {
    // flat shared-aperture address truncates to the LDS byte offset
    return (unsigned)(unsigned long long)p;
}

// ---------------------------------------------------------------------------
// Dense GEMM  H[n,64] = X[n,64] @ W[64,64]  via V_WMMA_F32_16X16X4_F32.
// TDM stages W (64x64) and a 128x64 X tile into LDS; each of the 8 waves
// computes a 16x64 output strip (4 column tiles), K=64 in steps of 4.
//
// A (16x4 f32): lanes 0-15 M=lane, lanes 16-31 M=lane-16;
//   VGPR0 K = 0+2*(lane>=16), VGPR1 K = 1+2*(lane>=16).
// B (4x16 f32): VGPR v holds row K = v + 2*(lane>=16), N = lane&15.
// C/D (16x16 f32): VGPR r holds row M = r + 8*(lane>=16), N = lane&15.
// ---------------------------------------------------------------------------
__global__ __launch_bounds__(256) void gemm64_wmma_kernel(
        const float* __restrict__ X, const float* __restrict__ W,
        float* __restrict__ H, int nrows) {
    __shared__ float sW[DIM * DIM];
    __shared__ float sX[ROWS_PER_BLOCK * DIM];

    const int lane = threadIdx.x & 31;
    const int wib  = threadIdx.x >> 5;                 // wave index in block, 0..7
    const int row0_block = blockIdx.x * ROWS_PER_BLOCK;

    if (wib == 0) {                                    // wave 0: EXEC all-ones
        // stage weights: full 64x64 tile
        tdm_load_2d(W, lds_offset(sW), DIM, DIM, DIM, DIM, DIM);
        // stage X rows [row0_block, row0_block+128); OOB rows zero-filled by TDM
        unsigned remaining = (unsigned)(nrows - row0_block);
        tdm_load_2d(X + (size_t)row0_block * DIM, lds_offset(sX),
                    DIM, remaining, DIM, ROWS_PER_BLOCK, DIM);
        __builtin_amdgcn_s_wait_tensorcnt(0);
    }
    __syncthreads();

    const int row0 = row0_block + wib * 16;
    if (row0 < nrows) {                                // uniform per wave
        const int m  = lane & 15;                      // row in tile (A) / col (B,C)
        const int kh = (lane >> 4) << 1;               // 0 or 2 : K offset per lane half
        const int mh = (lane >> 4) << 3;               // 0 or 8 : M offset for C/D rows

        v8f c0 = {}, c1 = {}, c2 = {}, c3 = {};
        const float* xr = sX + (wib * 16 + m) * DIM;

        #pragma unroll
        for (int k0 = 0; k0 < DIM; k0 += 4) {
            v2f a = *(const v2f*)(xr + k0 + kh);       // ds_load_b64
            const float* w0 = sW + (k0 + kh) * DIM + m;
            const float* w1 = sW + (k0 + kh + 1) * DIM + m;
            v2f b;
            b.x = w0[0];  b.y = w1[0];
            c0 = __builtin_amdgcn_wmma_f32_16x16x4_f32(false, a, false, b, (short)0, c0, false, false);
            b.x = w0[16]; b.y = w1[16];
            c1 = __builtin_amdgcn_wmma_f32_16x16x4_f32(false, a, false, b, (short)0, c1, false, false);
            b.x = w0[32]; b.y = w1[32];
            c2 = __builtin_amdgcn_wmma_f32_16x16x4_f32(false, a, false, b, (short)0, c2, false, false);
            b.x = w0[48]; b.y = w1[48];
            c3 = __builtin_amdgcn_wmma_f32_16x16x4_f32(false, a, false, b, (short)0, c3, false, false);
        }

        #pragma unroll
        for (int r = 0; r < 8; ++r) {
            float* outp = H + (size_t)(row0 + mh + r) * DIM + m;
            outp[0]  = c0[r];
            outp[16] = c1[r];
            outp[32] = c2[r];
            outp[48] = c3[r];
        }
    }
}

// ---------------------------------------------------------------------------
// Zero the aggregation buffer (float4 stores)
// ---------------------------------------------------------------------------
__global__ void zero_kernel(float4* __restrict__ p, int n4) {
    int i = blockIdx.x * blockDim.x + threadIdx.x;
    if (i < n4) p[i] = make_float4(0.f, 0.f, 0.f, 0.f);
}

// ---------------------------------------------------------------------------
// Edge scatter: agg[dst] += h[src] * dinv[src]*dinv[dst]
// 16 threads per edge, float4 gather + 4 scalar fp32 atomics (L2 atomic units).
// ---------------------------------------------------------------------------
__global__ void scatter_edges_kernel(const float* __restrict__ H,
                                     const int* __restrict__ src,
                                     const int* __restrict__ dst,
                                     const float* __restrict__ dinv,
                                     float* __restrict__ agg, int ne) {
    long long t = (long long)blockIdx.x * blockDim.x + threadIdx.x;
    int e = (int)(t >> 4);
    if (e >= ne) return;
    int f = ((int)t & 15) << 2;
    int s = src[e];
    int d = dst[e];
    float nrm = dinv[s] * dinv[d];
    float4 hv = *(const float4*)(H + (size_t)s * DIM + f);
    float* ap = agg + (size_t)d * DIM + f;
    atomicAdd(ap + 0, hv.x * nrm);
    atomicAdd(ap + 1, hv.y * nrm);
    atomicAdd(ap + 2, hv.z * nrm);
    atomicAdd(ap + 3, hv.w * nrm);
}

// ---------------------------------------------------------------------------
// Epilogue: out = relu(agg + h*dinv^2 (self-loop) + bias)
// ---------------------------------------------------------------------------
__global__ void epilogue_kernel(const float* __restrict__ agg,
                                const float* __restrict__ H,
                                const float* __restrict__ dinv,
                                const float* __restrict__ bias,
                                float* __restrict__ out, int n) {
    long long t = (long long)blockIdx.x * blockDim.x + threadIdx.x;
    int i = (int)(t >> 4);
    if (i >= n) return;
    int f = ((int)t & 15) << 2;
    float di = dinv[i];
    float self = di * di;
    float4 a  = *(const float4*)(agg + (size_t)i * DIM + f);
    float4 hv = *(const float4*)(H + (size_t)i * DIM + f);
    float4 bb = *(const float4*)(bias + f);
    float4 r;
    r.x = fmaxf(fmaf(hv.x, self, a.x) + bb.x, 0.f);
    r.y = fmaxf(fmaf(hv.y, self, a.y) + bb.y, 0.f);
    r.z = fmaxf(fmaf(hv.z, self, a.z) + bb.z, 0.f);
    r.w = fmaxf(fmaf(hv.w, self, a.w) + bb.w, 0.f);
    *(float4*)(out + (size_t)i * DIM + f) = r;
}

// ---------------------------------------------------------------------------
// Host-side orchestration
// ---------------------------------------------------------------------------
extern "C" void kernel_launch(void* const* d_in, const int* in_sizes, int n_in,
                              void* d_out, int out_size, void* d_ws, size_t ws_size,
                              hipStream_t stream) {
    const float* x    = (const float*)d_in[0];
    const int*   ei   = (const int*)d_in[1];
    const float* W1   = (const float*)d_in[2];
    const float* b1   = (const float*)d_in[3];
    const float* W2   = (const float*)d_in[4];
    const float* b2   = (const float*)d_in[5];
    float*       out  = (float*)d_out;

    const int N = in_sizes[0] / DIM;   // 50000
    const int E = in_sizes[1] / 2;     // 800000
    const int* src = ei;
    const int* dst = ei + E;

    // workspace layout (floats): deg[N] | dinv[N] | hbuf[N*64] | agg[N*64]
    float* deg  = (float*)d_ws;
    float* dinv = deg + N;
    float* hbuf = dinv + N;
    float* agg  = hbuf + (size_t)N * DIM;

    const int TB = 256;
    const int nblk_nodes = (N + TB - 1) / TB;
    const int nblk_edges = (E + TB - 1) / TB;
    const int nblk_gemm  = (N + ROWS_PER_BLOCK - 1) / ROWS_PER_BLOCK;
    const int n4         = N * (DIM / 4);
    const int nblk_zero  = (n4 + TB - 1) / TB;
    const long long scat_threads = (long long)E * 16;
    const int nblk_scat  = (int)((scat_threads + TB - 1) / TB);
    const long long epi_threads = (long long)N * 16;
    const int nblk_epi   = (int)((epi_threads + TB - 1) / TB);

    // ---- degree normalization ----
    deg_init_kernel<<<nblk_nodes, TB, 0, stream>>>(deg, N);
    deg_count_kernel<<<nblk_edges, TB, 0, stream>>>(dst, deg, E);
    deg_rsqrt_kernel<<<nblk_nodes, TB, 0, stream>>>(deg, dinv, N);

    // ---- layer 1: h = x @ W1 ; aggregate ; relu -> d_out ----
    gemm64_wmma_kernel<<<nblk_gemm, TB, 0, stream>>>(x, W1, hbuf, N);
    zero_kernel<<<nblk_zero, TB, 0, stream>>>((float4*)agg, n4);
    scatter_edges_kernel<<<nblk_scat, TB, 0, stream>>>(hbuf, src, dst, dinv, agg, E);
    epilogue_kernel<<<nblk_epi, TB, 0, stream>>>(agg, hbuf, dinv, b1, out, N);

    // ---- layer 2: h = out @ W2 ; aggregate ; relu -> d_out ----
    gemm64_wmma_kernel<<<nblk_gemm, TB, 0, stream>>>(out, W2, hbuf, N);
    zero_kernel<<<nblk_zero, TB, 0, stream>>>((float4*)agg, n4);
    scatter_edges_kernel<<<nblk_scat, TB, 0, stream>>>(hbuf, src, dst, dinv, agg, E);
    epilogue_kernel<<<nblk_epi, TB, 0, stream>>>(agg, hbuf, dinv, b2, out, N);
}